// WindowLayer_23897198035309
// MI455X (gfx1250) — compile-verified
//
#include <hip/hip_runtime.h>
#include <hip/hip_bf16.h>
#include <math.h>

#define B_SZ     1024
#define IN_F     1024
#define N_COMP   10
#define SEQ_LEN  2048
#define N_CHARS  80
#define THREE_N  30

typedef __attribute__((ext_vector_type(2))) float v2f;
typedef __attribute__((ext_vector_type(4))) float v4f;
typedef __attribute__((ext_vector_type(8))) float v8f;

// ---------------------------------------------------------------------------
// Kernel 0: initialize the "terminal" AND-flag to 1
// ---------------------------------------------------------------------------
__global__ void init_flag_kernel(int* __restrict__ flag) { *flag = 1; }

// ---------------------------------------------------------------------------
// Kernel 1: out = exp(inputs @ W^T + bias) via V_WMMA_F32_16X16X4_F32 (f32!)
// Grid: 64 blocks (16 rows each), 64 threads = 2 waves, wave w -> cols 16w..16w+15.
// A frag (16x4 f32): lanes 0-15 hold K=0,1 ; lanes 16-31 hold K=2,3.
// B frag (4x16 f32): VGPR0 = K0 (lanes0-15) / K2 (lanes16-31), VGPR1 = K1/K3.
// D (16x16 f32): VGPR v -> row v (lanes 0-15) / row v+8 (lanes 16-31), col = lane&15.
//
// Branch-free B load: out-of-range columns (30,31) clamp their address to
// column 29; the resulting garbage lands only in D columns that the epilogue
// never stores (col < 30 guard), since D column n depends solely on B column n.
// ---------------------------------------------------------------------------
__global__ __launch_bounds__(64)
void gemm_exp_kernel(const float* __restrict__ inputs,
                     const float* __restrict__ hidden,
                     const float* __restrict__ W,
                     const float* __restrict__ bias,
                     float* __restrict__ a_ws,
                     float* __restrict__ b_ws,
                     float* __restrict__ nh_out,
                     int*   __restrict__ flag)
{
    const int lane  = threadIdx.x & 31;
    const int wave  = threadIdx.x >> 5;
    const int nbase = wave * 16;
    const int mbase = blockIdx.x * 16;
    const int half  = lane >> 4;       // 0: lanes 0-15, 1: lanes 16-31
    const int l16   = lane & 15;
    const int koff  = half * 2;

    const int arow  = mbase + l16;                     // inputs row this lane feeds
    int       bcol  = nbase + l16;                     // W row (= output column)
    const int bclmp = bcol < THREE_N ? bcol : THREE_N - 1;  // clamp, no masking needed

    const float* aptr = inputs + (size_t)arow  * IN_F + koff;
    const float* bptr = W      + (size_t)bclmp * IN_F + koff;

    v8f acc = {};
    #pragma unroll 4
    for (int k = 0; k < IN_F; k += 4) {
        v2f afrag;
        afrag.x = aptr[k];
        afrag.y = aptr[k + 1];
        v2f bfrag;
        bfrag.x = bptr[k];
        bfrag.y = bptr[k + 1];
        acc = __builtin_amdgcn_wmma_f32_16x16x4_f32(
                  /*neg_a=*/false, afrag, /*neg_b=*/false, bfrag,
                  /*c_mod=*/(short)0, acc, /*reuse_a=*/false, /*reuse_b=*/false);
    }

    const int col = nbase + l16;
    if (col < THREE_N) {
        const float bv = bias[col];
        #pragma unroll
        for (int v = 0; v < 8; ++v) {
            const int   row = mbase + v + half * 8;
            const float val = expf(acc[v] + bv);
            if (col < N_COMP) {
                a_ws[row * N_COMP + col] = val;
            } else if (col < 2 * N_COMP) {
                b_ws[row * N_COMP + (col - N_COMP)] = val;
            } else {
                const int j  = col - 2 * N_COMP;
                const float nh = hidden[row * N_COMP + j] + val;
                nh_out[row * N_COMP + j] = nh;
                if (!(nh > (float)SEQ_LEN - 0.5f)) atomicAnd(flag, 0);
            }
        }
    }
}

// ---------------------------------------------------------------------------
// Kernel 2: phi (fused, LDS) + window = einsum('bs,bsc->bc', phi, sequence)
// One block per batch; 320 threads = 10 wave32.
// Streams 640 KB/batch of `sequence` exactly once (non-temporal).
// ---------------------------------------------------------------------------
__global__ __launch_bounds__(320)
void window_kernel(const float* __restrict__ sequence,
                   const float* __restrict__ a_ws,
                   const float* __restrict__ b_ws,
                   const float* __restrict__ nh,
                   const int*   __restrict__ flag,
                   float* __restrict__ window,
                   float* __restrict__ terminal)
{
    __shared__ float s_phi[SEQ_LEN];
    __shared__ float s_par[3 * N_COMP];
    __shared__ v4f   s_red[16][20];

    const int batch = blockIdx.x;
    const int tid   = threadIdx.x;

    if (batch == 0 && tid == 0) *terminal = (*flag) ? 1.0f : 0.0f;

    if (tid < N_COMP) {
        s_par[tid]              = a_ws[batch * N_COMP + tid];
        s_par[tid + N_COMP]     = b_ws[batch * N_COMP + tid];
        s_par[tid + 2 * N_COMP] = nh[batch * N_COMP + tid];
    }
    __syncthreads();

    float pa[N_COMP], pb[N_COMP], ph[N_COMP];
    #pragma unroll
    for (int j = 0; j < N_COMP; ++j) {
        pa[j] = s_par[j];
        pb[j] = s_par[j + N_COMP];
        ph[j] = s_par[j + 2 * N_COMP];
    }

    // Phase 1: phi[s] into LDS (v_exp_f32 path)
    for (int s = tid; s < SEQ_LEN; s += 320) {
        const float r = (float)s;
        float acc = 0.0f;
        #pragma unroll
        for (int j = 0; j < N_COMP; ++j) {
            const float d = ph[j] - r;
            acc += pa[j] * __expf(-pb[j] * d * d);
        }
        s_phi[s] = acc;
    }
    __syncthreads();

    // Phase 2: stream sequence once; fixed 4 columns per thread (no atomics)
    const int x  = tid % 20;   // column group: cols 4x..4x+3
    const int sy = tid / 20;   // 0..15
    v4f acc4 = {0.0f, 0.0f, 0.0f, 0.0f};
    const float* base = sequence + (size_t)batch * SEQ_LEN * N_CHARS + x * 4;
    #pragma unroll 4
    for (int k = 0; k < SEQ_LEN / 16; ++k) {
        const int   s = sy + (k << 4);
        const float p = s_phi[s];
        const v4f   v = __builtin_nontemporal_load(
                            (const v4f*)(base + (size_t)s * N_CHARS));
        acc4 += p * v;
    }

    // Phase 3: reduce the 16 sy-partials per column group
    s_red[sy][x] = acc4;
    __syncthreads();
    if (tid < 20) {
        v4f sum = s_red[0][tid];
        #pragma unroll
        for (int r = 1; r < 16; ++r) sum += s_red[r][tid];
        *(v4f*)(window + batch * N_CHARS + tid * 4) = sum;
    }
}

// ---------------------------------------------------------------------------
extern "C" void kernel_launch(void* const* d_in, const int* in_sizes, int n_in,
                              void* d_out, int out_size, void* d_ws, size_t ws_size,
                              hipStream_t stream) {
    const float* inputs   = (const float*)d_in[0];   // (1024,1024)
    const float* hidden   = (const float*)d_in[1];   // (1024,10)
    const float* sequence = (const float*)d_in[2];   // (1024,2048,80)
    const float* W        = (const float*)d_in[3];   // (30,1024)
    const float* bias     = (const float*)d_in[4];   // (30,)

    float* out      = (float*)d_out;
    float* window   = out;                              // 1024*80
    float* nh_out   = out + B_SZ * N_CHARS;             // 1024*10
    float* terminal = nh_out + B_SZ * N_COMP;           // 1

    float* a_ws = (float*)d_ws;                         // 1024*10
    float* b_ws = a_ws + B_SZ * N_COMP;                 // 1024*10
    int*   flag = (int*)(b_ws + B_SZ * N_COMP);

    init_flag_kernel<<<1, 1, 0, stream>>>(flag);
    gemm_exp_kernel<<<B_SZ / 16, 64, 0, stream>>>(inputs, hidden, W, bias,
                                                  a_ws, b_ws, nh_out, flag);
    window_kernel<<<B_SZ, 320, 0, stream>>>(sequence, a_ws, b_ws, nh_out,
                                            flag, window, terminal);
}